// SVNet_37692632990117
// MI455X (gfx1250) — compile-verified
//
#include <hip/hip_runtime.h>

typedef __attribute__((ext_vector_type(2))) float v2f;
typedef __attribute__((ext_vector_type(8))) float v8f;
typedef __attribute__((ext_vector_type(4))) int   v4i;

#define D 128            // feature dim (fixed by problem)
#define EPSBN 1e-5f

// ---------------------------------------------------------------------------
// CDNA5 async global->LDS DMA staging (ASYNCcnt-tracked). Falls back to a
// plain b128 copy if the builtin is unavailable on this toolchain.
// Builtin signature (from clang diagnostic): (v4i AS1*, v4i AS3*, imm, imm).
// ---------------------------------------------------------------------------
#if defined(__gfx1250__) && __has_builtin(__builtin_amdgcn_global_load_async_to_lds_b128)
#define SV_ASYNC_LDS 1
#endif

__device__ __forceinline__ void sv_stage16(const float* g, float* l) {
#ifdef SV_ASYNC_LDS
  __builtin_amdgcn_global_load_async_to_lds_b128(
      (__attribute__((address_space(1))) v4i*)(void*)g,
      (__attribute__((address_space(3))) v4i*)(void*)l,
      /*offset=*/0, /*cpol=*/0);
#else
  *(float4*)l = *(const float4*)g;
#endif
}

__device__ __forceinline__ void sv_stage_wait() {
#ifdef SV_ASYNC_LDS
#if __has_builtin(__builtin_amdgcn_s_wait_asynccnt)
  __builtin_amdgcn_s_wait_asynccnt(0);
#else
  asm volatile("s_wait_asynccnt 0" ::: "memory");
#endif
#endif
}

// ---------------------------------------------------------------------------
// 1) Edge scatter: one wave (32 lanes) per edge. Coalesced b128 read of the
//    source row; 4 float atomics per lane into the destination row.
//    agg[dst] += h[src]; cnt[dst] += 1
// ---------------------------------------------------------------------------
__global__ void svnet_scatter(const float* __restrict__ h,
                              const int* __restrict__ src,
                              const int* __restrict__ dst,
                              float* __restrict__ agg,
                              float* __restrict__ cnt,
                              int E) {
  int wave = (int)((blockIdx.x * blockDim.x + threadIdx.x) >> 5);
  int lane = threadIdx.x & 31;
  if (wave >= E) return;
  int s = src[wave];
  int t = dst[wave];
  const float4* hs = (const float4*)(h + (size_t)s * D);   // 32 lanes x 16B = row
  float* at = agg + (size_t)t * D;
  float4 v = hs[lane];
  atomicAdd(&at[lane * 4 + 0], v.x);
  atomicAdd(&at[lane * 4 + 1], v.y);
  atomicAdd(&at[lane * 4 + 2], v.z);
  atomicAdd(&at[lane * 4 + 3], v.w);
  if (lane == 0) atomicAdd(&cnt[t], 1.0f);
}

// ---------------------------------------------------------------------------
// 2) SAGE linear: out = (agg/cnt) @ Wl^T + bl + h @ Wr^T
//    Block = 256 threads = 8 waves; block owns a 16-row stripe, each wave owns
//    one 16-column tile. Tiles staged by async DMA to LDS; 1/cnt row-scale is
//    folded into the A operand at WMMA time. 64 x v_wmma_f32_16x16x4_f32.
// ---------------------------------------------------------------------------
__global__ void svnet_sage_gemm(const float* __restrict__ h,
                                const float* __restrict__ agg,
                                const float* __restrict__ cnt,
                                const float* __restrict__ Wl,  // [D][D] (out,k)
                                const float* __restrict__ bl,  // [D]
                                const float* __restrict__ Wr,  // [D][D]
                                float* __restrict__ out,
                                int N) {
  __shared__ float sA[16 * D];   // raw aggregate tile
  __shared__ float sH[16 * D];   // self features tile
  __shared__ float sInv[16];

  const int tid  = threadIdx.x;
  const int row0 = blockIdx.x * 16;

  if (tid < 16) {
    int r = row0 + tid;
    float c = (r < N) ? cnt[r] : 0.0f;
    sInv[tid] = (c > 0.0f) ? (1.0f / c) : 0.0f;
  }

  // Stage 16x128 agg + h tiles: 512 float4 slots each, 2 passes of 256 lanes.
#pragma unroll
  for (int p = 0; p < 2; ++p) {
    int idx = p * 256 + tid;      // float4 slot
    int r   = idx >> 5;           // 0..15
    int c4  = (idx & 31) * 4;     // 0,4,...,124
    int gr  = row0 + r;
    if (gr >= N) gr = N - 1;      // clamp (harmless duplicate reads)
    sv_stage16(agg + (size_t)gr * D + c4, &sA[r * D + c4]);
    sv_stage16(h   + (size_t)gr * D + c4, &sH[r * D + c4]);
  }
  sv_stage_wait();
  __syncthreads();

  const int wave    = tid >> 5;           // 0..7 -> column tile
  const int lane    = tid & 31;
  const int colBase = wave * 16;
  const int m       = lane & 15;          // A row / B,D column within tile
  const int khalf   = (lane >> 4) * 2;    // hi half-wave handles K+2,K+3
  const int ncol    = colBase + m;
  const float invm  = sInv[m];            // mean-normalization for A rows

  v8f acc = {};
#pragma unroll
  for (int k0 = 0; k0 < D; k0 += 4) {
    v2f a, b;
    // A: (agg * 1/cnt) tile, 16x4 fp32
    a.x = sA[m * D + k0 + khalf] * invm;
    a.y = sA[m * D + k0 + khalf + 1] * invm;
    // B: Wl^T tile -> B[k][n] = Wl[n][k]
    b.x = Wl[(size_t)ncol * D + k0 + khalf];
    b.y = Wl[(size_t)ncol * D + k0 + khalf + 1];
    acc = __builtin_amdgcn_wmma_f32_16x16x4_f32(false, a, false, b,
                                                (short)0, acc, false, false);
    a.x = sH[m * D + k0 + khalf];
    a.y = sH[m * D + k0 + khalf + 1];
    b.x = Wr[(size_t)ncol * D + k0 + khalf];
    b.y = Wr[(size_t)ncol * D + k0 + khalf + 1];
    acc = __builtin_amdgcn_wmma_f32_16x16x4_f32(false, a, false, b,
                                                (short)0, acc, false, false);
  }

  const float bias = bl[ncol];
  const int rhi = (lane >> 4) * 8;   // D layout: VGPR j holds M=j (lo) / j+8 (hi)
#pragma unroll
  for (int j = 0; j < 8; ++j) {
    int row = row0 + j + rhi;
    if (row < N) out[(size_t)row * D + ncol] = acc[j] + bias;
  }
}

// ---------------------------------------------------------------------------
// 3) Single-input GEMM for readout: out = in @ W^T + b
// ---------------------------------------------------------------------------
__global__ void svnet_gemm1(const float* __restrict__ in,
                            const float* __restrict__ W,  // [D][D]
                            const float* __restrict__ b,  // [D]
                            float* __restrict__ out,
                            int N) {
  __shared__ float sA[16 * D];
  const int tid  = threadIdx.x;
  const int row0 = blockIdx.x * 16;

#pragma unroll
  for (int p = 0; p < 2; ++p) {
    int idx = p * 256 + tid;
    int r   = idx >> 5;
    int c4  = (idx & 31) * 4;
    int gr  = row0 + r;
    if (gr >= N) gr = N - 1;
    sv_stage16(in + (size_t)gr * D + c4, &sA[r * D + c4]);
  }
  sv_stage_wait();
  __syncthreads();

  const int wave    = tid >> 5;
  const int lane    = tid & 31;
  const int colBase = wave * 16;
  const int m       = lane & 15;
  const int khalf   = (lane >> 4) * 2;
  const int ncol    = colBase + m;

  v8f acc = {};
#pragma unroll
  for (int k0 = 0; k0 < D; k0 += 4) {
    v2f a, bb;
    a.x  = sA[m * D + k0 + khalf];
    a.y  = sA[m * D + k0 + khalf + 1];
    bb.x = W[(size_t)ncol * D + k0 + khalf];
    bb.y = W[(size_t)ncol * D + k0 + khalf + 1];
    acc = __builtin_amdgcn_wmma_f32_16x16x4_f32(false, a, false, bb,
                                                (short)0, acc, false, false);
  }

  const float bias = b[ncol];
  const int rhi = (lane >> 4) * 8;
#pragma unroll
  for (int j = 0; j < 8; ++j) {
    int row = row0 + j + rhi;
    if (row < N) out[(size_t)row * D + ncol] = acc[j] + bias;
  }
}

// ---------------------------------------------------------------------------
// 4a) BN column partial sums, coalesced: each wave reads whole 128-float rows
//     as float4 per lane; per-lane channel partials; LDS cross-wave combine;
//     atomicAdd into global accumulators (sumAcc[D], sqAcc[D]).
// ---------------------------------------------------------------------------
__global__ void svnet_colsum(const float* __restrict__ t, int N,
                             float* __restrict__ sumAcc,
                             float* __restrict__ sqAcc) {
  __shared__ float s1[8 * D];
  __shared__ float s2[8 * D];
  const int lane = threadIdx.x & 31;
  const int wv   = threadIdx.x >> 5;       // 0..7
  float4 s = {0.f, 0.f, 0.f, 0.f};
  float4 q = {0.f, 0.f, 0.f, 0.f};
  for (int r = blockIdx.x * 8 + wv; r < N; r += gridDim.x * 8) {
    float4 v = *(const float4*)&t[(size_t)r * D + lane * 4];
    s.x += v.x; s.y += v.y; s.z += v.z; s.w += v.w;
    q.x += v.x * v.x; q.y += v.y * v.y; q.z += v.z * v.z; q.w += v.w * v.w;
  }
  s1[wv * D + lane * 4 + 0] = s.x; s1[wv * D + lane * 4 + 1] = s.y;
  s1[wv * D + lane * 4 + 2] = s.z; s1[wv * D + lane * 4 + 3] = s.w;
  s2[wv * D + lane * 4 + 0] = q.x; s2[wv * D + lane * 4 + 1] = q.y;
  s2[wv * D + lane * 4 + 2] = q.z; s2[wv * D + lane * 4 + 3] = q.w;
  __syncthreads();
  if (threadIdx.x < D) {
    float a = 0.f, b = 0.f;
#pragma unroll
    for (int w = 0; w < 8; ++w) {
      a += s1[w * D + threadIdx.x];
      b += s2[w * D + threadIdx.x];
    }
    atomicAdd(&sumAcc[threadIdx.x], a);
    atomicAdd(&sqAcc[threadIdx.x], b);
  }
}

// 4b) finalize: mean = S/N, rstd = rsqrt(S2/N - mean^2 + eps)  (biased var)
__global__ void svnet_bn_finalize(const float* __restrict__ sumAcc,
                                  const float* __restrict__ sqAcc, int N,
                                  float* __restrict__ mean,
                                  float* __restrict__ rstd) {
  int c = threadIdx.x;
  float invN = 1.0f / (float)N;
  float mu = sumAcc[c] * invN;
  float var = sqAcc[c] * invN - mu * mu;
  mean[c] = mu;
  rstd[c] = rsqrtf(var + EPSBN);
}

// ---------------------------------------------------------------------------
// 5) BN apply + ReLU, float4-vectorized over N*D
// ---------------------------------------------------------------------------
__global__ void svnet_bn_relu4(const float4* __restrict__ t,
                               const float* __restrict__ mean,
                               const float* __restrict__ rstd,
                               const float* __restrict__ gamma,
                               const float* __restrict__ beta,
                               float4* __restrict__ out, int total4) {
  int i = blockIdx.x * blockDim.x + threadIdx.x;
  if (i >= total4) return;
  int c = (i << 2) & (D - 1);
  float4 v = t[i];
  float4 r;
  r.x = fmaxf((v.x - mean[c + 0]) * rstd[c + 0] * gamma[c + 0] + beta[c + 0], 0.f);
  r.y = fmaxf((v.y - mean[c + 1]) * rstd[c + 1] * gamma[c + 1] + beta[c + 1], 0.f);
  r.z = fmaxf((v.z - mean[c + 2]) * rstd[c + 2] * gamma[c + 2] + beta[c + 2], 0.f);
  r.w = fmaxf((v.w - mean[c + 3]) * rstd[c + 3] * gamma[c + 3] + beta[c + 3], 0.f);
  out[i] = r;
}

// ---------------------------------------------------------------------------
// 6) atom = t * atom_mask[row], float4-vectorized (bias already in t)
// ---------------------------------------------------------------------------
__global__ void svnet_mask4(const float4* __restrict__ t,
                            const float* __restrict__ mask,
                            float4* __restrict__ out, int total4) {
  int i = blockIdx.x * blockDim.x + threadIdx.x;
  if (i >= total4) return;
  int r = i >> 5;                 // (i*4) / 128
  float m = mask[r];
  float4 v = t[i];
  v.x *= m; v.y *= m; v.z *= m; v.w *= m;
  out[i] = v;
}

// ---------------------------------------------------------------------------
// 7) state = h[u_index]  (one block per u, 32 lanes x float4)
// ---------------------------------------------------------------------------
__global__ void svnet_gather(const float* __restrict__ h,
                             const int* __restrict__ uidx,
                             float* __restrict__ out, int U) {
  int u = blockIdx.x;
  if (u >= U) return;
  int lane = threadIdx.x;         // 0..31
  const float4* src = (const float4*)(h + (size_t)uidx[u] * D);
  float4* dst = (float4*)(out + (size_t)u * D);
  dst[lane] = src[lane];
}

// ---------------------------------------------------------------------------
// Host-side orchestration
// ---------------------------------------------------------------------------
extern "C" void kernel_launch(void* const* d_in, const int* in_sizes, int n_in,
                              void* d_out, int out_size, void* d_ws, size_t ws_size,
                              hipStream_t stream) {
  const float* x     = (const float*)d_in[0];
  const int*   ei    = (const int*)d_in[1];
  const float* amask = (const float*)d_in[2];
  const int*   uidx  = (const int*)d_in[3];
  const float* llw   = (const float*)d_in[4];
  const float* llb   = (const float*)d_in[5];
  const float* lrw   = (const float*)d_in[6];
  const float* bng   = (const float*)d_in[7];
  const float* bnb   = (const float*)d_in[8];
  const float* row1  = (const float*)d_in[9];
  const float* rob1  = (const float*)d_in[10];
  const float* robng = (const float*)d_in[11];
  const float* robnb = (const float*)d_in[12];
  const float* row2  = (const float*)d_in[13];
  const float* rob2  = (const float*)d_in[14];

  const int N = in_sizes[0] / D;
  const int E = in_sizes[1] / 2;
  const int U = in_sizes[3];
  const int L = in_sizes[5] / D;

  const int* src = ei;
  const int* dst = ei + E;

  // workspace carve-up (floats)
  float* agg    = (float*)d_ws;               // N*D (reused as z buffer later)
  float* tmp    = agg + (size_t)N * D;        // N*D
  float* hbuf   = tmp + (size_t)N * D;        // N*D
  float* cnt    = hbuf + (size_t)N * D;       // N
  float* sumAcc = cnt + N;                    // D  (contiguous with sqAcc)
  float* sqAcc  = sumAcc + D;                 // D
  float* mean   = sqAcc + D;                  // D
  float* rstd   = mean + D;                   // D

  // output carve-up
  float* h_out     = (float*)d_out;            // [N, D]
  float* atom_out  = h_out + (size_t)N * D;    // [N, D]
  float* state_out = atom_out + (size_t)N * D; // [U, D]

  const int total    = N * D;
  const int total4   = total / 4;
  const int ewaves   = (E * 32 + 255) / 256;
  const int gemmBlks = (N + 15) / 16;
  const int ew4Blks  = (total4 + 255) / 256;
  const int csBlks   = 256;

  // ---- 3 SAGE layers ----
  for (int l = 0; l < L; ++l) {
    const float* hin  = (l == 0) ? x : hbuf;
    float*       hout = (l == L - 1) ? h_out : hbuf;

    (void)hipMemsetAsync(agg, 0, (size_t)N * D * sizeof(float), stream);
    (void)hipMemsetAsync(cnt, 0, (size_t)N * sizeof(float), stream);
    (void)hipMemsetAsync(sumAcc, 0, 2 * D * sizeof(float), stream);

    svnet_scatter<<<ewaves, 256, 0, stream>>>(hin, src, dst, agg, cnt, E);

    svnet_sage_gemm<<<gemmBlks, 256, 0, stream>>>(
        hin, agg, cnt,
        llw + (size_t)l * D * D, llb + (size_t)l * D,
        lrw + (size_t)l * D * D, tmp, N);

    svnet_colsum<<<csBlks, 256, 0, stream>>>(tmp, N, sumAcc, sqAcc);
    svnet_bn_finalize<<<1, D, 0, stream>>>(sumAcc, sqAcc, N, mean, rstd);

    svnet_bn_relu4<<<ew4Blks, 256, 0, stream>>>(
        (const float4*)tmp, mean, rstd,
        bng + (size_t)l * D, bnb + (size_t)l * D, (float4*)hout, total4);
  }

  // ---- readout ----
  // state = h[u_index]
  svnet_gather<<<U, 32, 0, stream>>>(h_out, uidx, state_out, U);

  // z = relu(bn(h @ ro_w1^T + ro_b1))   (z stored in agg buffer)
  (void)hipMemsetAsync(sumAcc, 0, 2 * D * sizeof(float), stream);
  svnet_gemm1<<<gemmBlks, 256, 0, stream>>>(h_out, row1, rob1, tmp, N);
  svnet_colsum<<<csBlks, 256, 0, stream>>>(tmp, N, sumAcc, sqAcc);
  svnet_bn_finalize<<<1, D, 0, stream>>>(sumAcc, sqAcc, N, mean, rstd);
  svnet_bn_relu4<<<ew4Blks, 256, 0, stream>>>(
      (const float4*)tmp, mean, rstd, robng, robnb, (float4*)agg, total4);

  // atom = (z @ ro_w2^T + ro_b2) * atom_mask
  svnet_gemm1<<<gemmBlks, 256, 0, stream>>>(agg, row2, rob2, tmp, N);
  svnet_mask4<<<ew4Blks, 256, 0, stream>>>(
      (const float4*)tmp, amask, (float4*)atom_out, total4);
}